// RankingConsistencyLoss_42013370090243
// MI455X (gfx1250) — compile-verified
//
#include <hip/hip_runtime.h>
#include <hip/hip_bf16.h>
#include <stdint.h>

// Problem constants (match the reference).
#define N_TOTAL 4194304
#define NUM_H   131072
#define CMAX    256            // LDS-staged group-size cap (Poisson(32) tail << 256)
#define WAVES_PER_BLOCK 8
#define SCAN_BLOCKS (NUM_H / 256)   // 512

// CDNA5 async global->LDS builtins (ASYNCcnt-tracked).
// Probe-confirmed prototypes (from hipcc diagnostics):
//   b32 : (int* gsrc, int* lds_dst, imm offset, imm cpol)
//   b128: (v4i* gsrc, v4i* lds_dst, imm offset, imm cpol), v4i = int vector_size(16)
typedef int v4i __attribute__((vector_size(16)));

#if defined(__HIP_DEVICE_COMPILE__) && __has_builtin(__builtin_amdgcn_global_load_async_to_lds_b128)
#define ASYNC_LDS_B128 1
#else
#define ASYNC_LDS_B128 0
#endif
#if defined(__HIP_DEVICE_COMPILE__) && __has_builtin(__builtin_amdgcn_global_load_async_to_lds_b32)
#define ASYNC_LDS_B32 1
#else
#define ASYNC_LDS_B32 0
#endif

// Sortable-key transform: unsigned compare of key == float compare, with
// original index in the low 32 bits as the stable tie-break (== jnp.lexsort).
__device__ __forceinline__ unsigned sortkey32(float f) {
  unsigned u = __float_as_uint(f);
  return u ^ ((unsigned)(((int)u) >> 31) | 0x80000000u);
}

// ---------------------------------------------------------------------------
// One wave32 per group (avg group size == 32 == wave width). rank_i == number
// of j with K_j < K_i (strict total order, idx tie-break). The only
// data-dependent reduction is the exact integer S = sum(rank_p * rank_t);
// sum(r) and sum((r-mu)^2) have closed forms, so the result is invariant to
// the nondeterministic scatter order.
// ---------------------------------------------------------------------------
__global__ __launch_bounds__(256) void k_group(
    const int* __restrict__ starts, const int* __restrict__ counts,
    ulonglong2* __restrict__ gk,
    double* __restrict__ total, unsigned int* __restrict__ cnt) {
  __shared__ ulonglong2 sk[WAVES_PER_BLOCK * CMAX];   // 32 KB

  const int wave = threadIdx.x >> 5;
  const int lane = threadIdx.x & 31;
  const int g = blockIdx.x * WAVES_PER_BLOCK + wave;
  const int c = counts[g];
  if (c < 2) return;  // matches valid = counts >= 2 (c uniform per wave; no barriers below)
  const int start = starts[g];

  long long S = 0;
  if (__builtin_expect(c <= CMAX, 1)) {
    ulonglong2* lk = sk + wave * CMAX;
    for (int k = lane; k < c; k += 32) {
#if ASYNC_LDS_B128
      __builtin_amdgcn_global_load_async_to_lds_b128((v4i*)(gk + start + k), (v4i*)(lk + k), 0, 0);
#elif ASYNC_LDS_B32
      int* gsrc = (int*)(gk + start + k);
      int* ldst = (int*)(lk + k);
      __builtin_amdgcn_global_load_async_to_lds_b32(gsrc + 0, ldst + 0, 0, 0);
      __builtin_amdgcn_global_load_async_to_lds_b32(gsrc + 1, ldst + 1, 0, 0);
      __builtin_amdgcn_global_load_async_to_lds_b32(gsrc + 2, ldst + 2, 0, 0);
      __builtin_amdgcn_global_load_async_to_lds_b32(gsrc + 3, ldst + 3, 0, 0);
#else
      lk[k] = gk[start + k];
#endif
    }
#if ASYNC_LDS_B128 || ASYNC_LDS_B32
#if __has_builtin(__builtin_amdgcn_s_wait_asynccnt)
    __builtin_amdgcn_s_wait_asynccnt(0);
#else
    asm volatile("s_wait_asynccnt 0" ::: "memory");
#endif
    asm volatile("" ::: "memory");   // keep LDS reads after the wait
#endif
    for (int i = lane; i < c; i += 32) {
      const ulonglong2 Ki = lk[i];
      int rp = 0, rt = 0;
      for (int j = 0; j < c; j++) {   // all lanes read same j -> ds_load_b128 broadcast
        const ulonglong2 Kj = lk[j];
        rp += (Kj.x < Ki.x);
        rt += (Kj.y < Ki.y);
      }
      S += (long long)rp * (long long)rt;
    }
  } else {  // defensive fallback for huge groups: loop straight from L2
    const ulonglong2* lk = gk + start;
    for (int i = lane; i < c; i += 32) {
      const ulonglong2 Ki = lk[i];
      long long rp = 0, rt = 0;
      for (int j = 0; j < c; j++) {
        const ulonglong2 Kj = lk[j];
        rp += (Kj.x < Ki.x);
        rt += (Kj.y < Ki.y);
      }
      S += rp * rt;
    }
  }

  for (int off = 16; off > 0; off >>= 1) S += __shfl_xor(S, off, 32);

  if (lane == 0) {
    const double dc  = (double)c;
    const double mu  = (dc - 1.0) * 0.5;
    const double var = dc * (dc * dc - 1.0) / 12.0;  // = sum((r-mu)^2), same for a and b
    double na = sqrt(var);
    na = (na > 1e-8) ? na : 1e-8;                    // matches max(norm, EPS)
    const double corr = ((double)S - dc * mu * mu) / (na * na);
    atomicAdd(total, 1.0 - corr);
    atomicAdd(cnt, 1u);
  }
}

__global__ void k_zero(int* __restrict__ counts, double* __restrict__ total,
                       unsigned int* __restrict__ cnt) {
  int i = blockIdx.x * blockDim.x + threadIdx.x;
  if (i < NUM_H) counts[i] = 0;
  if (i == 0) { *total = 0.0; *cnt = 0u; }
}

// Histogram: 4 ids per thread via B128 loads (HBM-bound pass), L2 atomics.
__global__ void k_hist(const int4* __restrict__ hid4, int* __restrict__ counts) {
  int i = blockIdx.x * blockDim.x + threadIdx.x;  // i indexes int4 chunks
  if (i < N_TOTAL / 4) {
    if (i + 2048 < N_TOTAL / 4) __builtin_prefetch(&hid4[i + 2048], 0, 0);  // global_prefetch_b8
    int4 h = hid4[i];
    atomicAdd(&counts[h.x], 1);
    atomicAdd(&counts[h.y], 1);
    atomicAdd(&counts[h.z], 1);
    atomicAdd(&counts[h.w], 1);
  }
}

// --------- 3-phase fully parallel exclusive scan over 128K bins -------------
__global__ __launch_bounds__(256) void k_scan1(const int* __restrict__ counts,
                                               int* __restrict__ blockSums) {
  __shared__ int sh[256];
  const int t = threadIdx.x;
  sh[t] = counts[blockIdx.x * 256 + t];
  __syncthreads();
  for (int off = 128; off > 0; off >>= 1) {
    if (t < off) sh[t] += sh[t + off];
    __syncthreads();
  }
  if (t == 0) blockSums[blockIdx.x] = sh[0];
}

__global__ __launch_bounds__(512) void k_scan2(int* __restrict__ blockSums) {
  __shared__ int sh[SCAN_BLOCKS];
  const int t = threadIdx.x;
  const int v = blockSums[t];
  sh[t] = v;
  __syncthreads();
  for (int off = 1; off < SCAN_BLOCKS; off <<= 1) {
    int u = (t >= off) ? sh[t - off] : 0;
    __syncthreads();
    sh[t] += u;
    __syncthreads();
  }
  blockSums[t] = sh[t] - v;  // exclusive block offset
}

__global__ __launch_bounds__(256) void k_scan3(const int* __restrict__ counts,
                                               const int* __restrict__ blockSums,
                                               int* __restrict__ starts,
                                               int* __restrict__ cursor) {
  __shared__ int sh[256];
  const int t = threadIdx.x;
  const int i = blockIdx.x * 256 + t;
  const int v = counts[i];
  sh[t] = v;
  __syncthreads();
  for (int off = 1; off < 256; off <<= 1) {
    int u = (t >= off) ? sh[t - off] : 0;
    __syncthreads();
    sh[t] += u;
    __syncthreads();
  }
  const int ex = sh[t] - v + blockSums[blockIdx.x];
  starts[i] = ex;
  cursor[i] = ex;
}

// Scatter into grouped key layout: 4 elements per thread with vector reads;
// each element becomes ONE 16B packed-key store (global_store_b128) into the
// L2-resident grouped array.
__global__ void k_scatter(const float4* __restrict__ pred4, const float4* __restrict__ targ4,
                          const int4* __restrict__ hid4, int* __restrict__ cursor,
                          ulonglong2* __restrict__ gk) {
  int i = blockIdx.x * blockDim.x + threadIdx.x;  // i indexes 4-element chunks
  if (i < N_TOTAL / 4) {
    const int4   h = hid4[i];
    const float4 p = pred4[i];
    const float4 q = targ4[i];
    const unsigned base = (unsigned)i * 4u;
    int pos;
    pos = atomicAdd(&cursor[h.x], 1);
    gk[pos] = make_ulonglong2(((unsigned long long)sortkey32(p.x) << 32) | (base + 0u),
                              ((unsigned long long)sortkey32(q.x) << 32) | (base + 0u));
    pos = atomicAdd(&cursor[h.y], 1);
    gk[pos] = make_ulonglong2(((unsigned long long)sortkey32(p.y) << 32) | (base + 1u),
                              ((unsigned long long)sortkey32(q.y) << 32) | (base + 1u));
    pos = atomicAdd(&cursor[h.z], 1);
    gk[pos] = make_ulonglong2(((unsigned long long)sortkey32(p.z) << 32) | (base + 2u),
                              ((unsigned long long)sortkey32(q.z) << 32) | (base + 2u));
    pos = atomicAdd(&cursor[h.w], 1);
    gk[pos] = make_ulonglong2(((unsigned long long)sortkey32(p.w) << 32) | (base + 3u),
                              ((unsigned long long)sortkey32(q.w) << 32) | (base + 3u));
  }
}

__global__ void k_final(const double* __restrict__ total,
                        const unsigned int* __restrict__ cnt,
                        float* __restrict__ out) {
  if (threadIdx.x == 0 && blockIdx.x == 0) {
    unsigned int c = *cnt;
    out[0] = (c > 0u) ? (float)(*total / (double)c) : 0.0f;
  }
}

extern "C" void kernel_launch(void* const* d_in, const int* in_sizes, int n_in,
                              void* d_out, int out_size, void* d_ws, size_t ws_size,
                              hipStream_t stream) {
  const float* pred = (const float*)d_in[0];
  const float* targ = (const float*)d_in[1];
  const int*   hid  = (const int*)d_in[2];
  // d_in[3] (env_ids) is unused by the reference loss.

  // Workspace layout (~66 MB; intermediates are L2-resident on MI455X 192MB L2):
  //  [0,512K)          counts      i32[H]
  //  [512K,1M)         starts      i32[H]
  //  [1M,1.5M)         cursor      i32[H]
  //  [1.5M,+68B)       accumulators (f64 total, u32 n_pairs)
  //  [1.5M+4K,+2K)     blockSums   i32[512]
  //  [2M,66M)          grouped packed keys  ulonglong2[N] (16B each)
  char* ws = (char*)d_ws;
  int*          counts    = (int*)(ws + 0);
  int*          starts    = (int*)(ws + (size_t)NUM_H * 4);
  int*          cursor    = (int*)(ws + (size_t)NUM_H * 8);
  double*       total     = (double*)(ws + (size_t)NUM_H * 12);
  unsigned int* cnt       = (unsigned int*)(ws + (size_t)NUM_H * 12 + 64);
  int*          blockSums = (int*)(ws + (size_t)NUM_H * 12 + 4096);
  ulonglong2*   gk        = (ulonglong2*)(ws + ((size_t)2u << 20));

  float* out = (float*)d_out;

  k_zero   <<<NUM_H / 256,       256, 0, stream>>>(counts, total, cnt);
  k_hist   <<<N_TOTAL / 4 / 256, 256, 0, stream>>>((const int4*)hid, counts);
  k_scan1  <<<SCAN_BLOCKS,       256, 0, stream>>>(counts, blockSums);
  k_scan2  <<<1,         SCAN_BLOCKS, 0, stream>>>(blockSums);
  k_scan3  <<<SCAN_BLOCKS,       256, 0, stream>>>(counts, blockSums, starts, cursor);
  k_scatter<<<N_TOTAL / 4 / 256, 256, 0, stream>>>((const float4*)pred, (const float4*)targ,
                                                   (const int4*)hid, cursor, gk);
  k_group  <<<NUM_H / WAVES_PER_BLOCK, 256, 0, stream>>>(starts, counts, gk, total, cnt);
  k_final  <<<1,                   1, 0, stream>>>(total, cnt, out);
}